// HNPLinear_25443386261609
// MI455X (gfx1250) — compile-verified
//
#include <hip/hip_runtime.h>

typedef float v2f __attribute__((ext_vector_type(2)));
typedef float v8f __attribute__((ext_vector_type(8)));

// Problem constants
#define BQ    8
#define CQ    32
#define D1Q   512
#define D2Q   512
#define NRCQ  1344

// Output offsets (floats): (zw0, zw1, zw2, u0, u1, u2)
#define ZW0_OFF 0L
#define ZW1_OFF 2097152L
#define ZW2_OFF 69206016L
#define U0_OFF  70516736L
#define U1_OFF  70647808L
#define U2_OFF  70778880L

// Workspace offsets (floats)
#define WS_RC       0L
#define WS_ADD0     10752L
#define WS_ADDB0    14848L
#define WS_ADDL1    15104L
#define WS_ADDBIAS1 15360L
#define WS_ADDFIN   15616L
#define WS_ROWV     18176L
#define WS_COLV     149248L
#define WS_ROWCAT   280320L
#define WS_COLCAT   2770688L

// rowcat row j -> original row_b index (column of W_l1_r)
__device__ __forceinline__ int map_row(int j) {
  return (j < 512) ? (96 + j)
       : (j < 544) ? (j - 512)     // rm1
       : (j < 576) ? (j - 480)     // b0: 64+(j-544)
                   : (j - 544);    // cm0: 32+(j-576)
}
// colcat row j -> original col_b index (column of W_l1_c / W_bias1)
__device__ __forceinline__ int map_col(int j) {
  return (j < 320) ? (96 + j) : (j - 320);
}

// One 16x16 fp32 tile via V_WMMA_F32_16X16X4_F32.
// A: M x K row-major (lda), B: K x N row-major (row stride ldb elements).
// PERM: 0 none, 1 map_row on A columns, 2 map_col on A columns.
template <int PERM>
__device__ __forceinline__ v8f wmma_tile_f32(const float* __restrict__ A, int lda,
                                             const float* __restrict__ Bm, long ldb,
                                             int K, int mbase, int nbase, int lane) {
  v8f acc = {0.f, 0.f, 0.f, 0.f, 0.f, 0.f, 0.f, 0.f};
  const int l15 = lane & 15;
  const int kh  = (lane >> 4) << 1;              // 0 or 2
  const float* arow = A + (long)(mbase + l15) * lda;
  const float* bcol = Bm + (nbase + l15);
  for (int k = 0; k < K; k += 4) {
    int ka = k + kh;
    int c0 = ka, c1 = ka + 1;
    if (PERM == 1) { c0 = map_row(c0); c1 = map_row(c1); }
    if (PERM == 2) { c0 = map_col(c0); c1 = map_col(c1); }
    v2f a, b;
    a.x = arow[c0];
    a.y = arow[c1];
    b.x = bcol[(long)ka * ldb];
    b.y = bcol[(long)(ka + 1) * ldb];
    acc = __builtin_amdgcn_wmma_f32_16x16x4_f32(false, a, false, b, (short)0, acc,
                                                false, false);
  }
  return acc;
}

// ---------------- Phase 1: reductions / assembly ----------------

__device__ __forceinline__ float blockReduce256(float v, float* red) {
  int t = threadIdx.x;
  red[t] = v;
  __syncthreads();
#pragma unroll
  for (int s = 128; s > 0; s >>= 1) {
    if (t < s) red[t] += red[t + s];
    __syncthreads();
  }
  float r = red[0];
  __syncthreads();
  return r;
}

// rowcat[b] (608x512): rows 0..511 = w0^T, 512..543 = rm1, 544..575 = b0, 576..607 = cm0
__global__ void k_rowcat(const float* __restrict__ w0, const float* __restrict__ w1,
                         const float* __restrict__ b0, float* __restrict__ ws) {
  int tid = blockIdx.x * blockDim.x + threadIdx.x;   // 131072 = B*C*D1
  int d = tid & 511, c = (tid >> 9) & 31, b = tid >> 14;
  float* rowcat = ws + WS_ROWCAT + (long)b * 608 * 512;
  const float* w1p = w1 + (long)(b * 32 + c) * 512 * 512 + d;
  float s = 0.f;
  for (int h = 0; h < 512; ++h) s += w1p[(long)h * 512];
  rowcat[(512 + c) * 512 + d] = s * (1.f / 512.f);
  const float* w0p = w0 + ((long)(b * 32 + c) * 512 + d) * 16;
  float s2 = 0.f;
#pragma unroll
  for (int i = 0; i < 16; ++i) {
    float t = w0p[i];
    s2 += t;
    rowcat[(c * 16 + i) * 512 + d] = t;
  }
  rowcat[(576 + c) * 512 + d] = s2 * (1.f / 16.f);
  rowcat[(544 + c) * 512 + d] = b0[(b * 32 + c) * 512 + d];
}

// colcat[b] (416x512): rows 0..319 = w2flat, 320..351 = cm1, 352..383 = b1, 384..415 = rm2
__global__ void k_colcat(const float* __restrict__ w1, const float* __restrict__ w2,
                         const float* __restrict__ b1, float* __restrict__ ws) {
  int tid = blockIdx.x * blockDim.x + threadIdx.x;   // 131072 = B*C*D2
  int d = tid & 511, c = (tid >> 9) & 31, b = tid >> 14;
  float* colcat = ws + WS_COLCAT + (long)b * 416 * 512;
  const float* w1p = w1 + ((long)(b * 32 + c) * 512 + d) * 512;
  float s = 0.f;
  for (int i = 0; i < 512; ++i) s += w1p[i];
  colcat[(320 + c) * 512 + d] = s * (1.f / 512.f);
  const float* w2p = w2 + (long)(b * 32 + c) * 10 * 512 + d;
  float s2 = 0.f;
#pragma unroll
  for (int no = 0; no < 10; ++no) {
    float t = w2p[no * 512];
    s2 += t;
    colcat[(c * 10 + no) * 512 + d] = t;
  }
  colcat[(384 + c) * 512 + d] = s2 * (1.f / 10.f);
  colcat[(352 + c) * 512 + d] = b1[(b * 32 + c) * 512 + d];
}

// rc[b, 1344] = [w0m(32)|w1m(32)|w2m(32)|b0m(32)|b1m(32)|b2m(32)|rm0(512)|cm2(320)|b2(320)]
__global__ void k_rc(const float* __restrict__ w0, const float* __restrict__ w1,
                     const float* __restrict__ w2, const float* __restrict__ b0,
                     const float* __restrict__ b1, const float* __restrict__ b2,
                     float* __restrict__ ws) {
  __shared__ float red[256];
  int c = blockIdx.x & 31, b = blockIdx.x >> 5, t = threadIdx.x;
  float* rc = ws + WS_RC + (long)b * NRCQ;
  const float* w1p = w1 + (long)(b * 32 + c) * 512 * 512;
  const float* w0p = w0 + (long)(b * 32 + c) * 512 * 16;
  const float* w2p = w2 + (long)(b * 32 + c) * 10 * 512;
  const float* b0p = b0 + (b * 32 + c) * 512;
  const float* b1p = b1 + (b * 32 + c) * 512;
  const float* b2p = b2 + (b * 32 + c) * 10;
  float s = 0.f;
  for (int i = t; i < 262144; i += 256) s += w1p[i];
  float sw1 = blockReduce256(s, red);
  s = 0.f;
  for (int i = t; i < 8192; i += 256) s += w0p[i];
  float sw0 = blockReduce256(s, red);
  s = 0.f;
  for (int i = t; i < 5120; i += 256) s += w2p[i];
  float sw2 = blockReduce256(s, red);
  s = 0.f;
  for (int i = t; i < 512; i += 256) s += b0p[i];
  float sb0 = blockReduce256(s, red);
  s = 0.f;
  for (int i = t; i < 512; i += 256) s += b1p[i];
  float sb1 = blockReduce256(s, red);
  s = (t < 10) ? b2p[t] : 0.f;
  float sb2 = blockReduce256(s, red);
  if (t == 0) {
    rc[c]       = sw0 * (1.f / 8192.f);
    rc[32 + c]  = sw1 * (1.f / 262144.f);
    rc[64 + c]  = sw2 * (1.f / 5120.f);
    rc[96 + c]  = sb0 * (1.f / 512.f);
    rc[128 + c] = sb1 * (1.f / 512.f);
    rc[160 + c] = sb2 * (1.f / 10.f);
  }
  if (t < 16) {  // rm0: mean over D1 of w0 column t
    float r0 = 0.f;
    for (int dd = 0; dd < 512; ++dd) r0 += w0p[dd * 16 + t];
    rc[192 + c * 16 + t] = r0 * (1.f / 512.f);
  }
  if (t < 10) {  // cm2 + b2 copy
    float cc = 0.f;
    const float* p = w2p + t * 512;
    for (int dd = 0; dd < 512; ++dd) cc += p[dd];
    rc[704 + c * 10 + t]  = cc * (1.f / 512.f);
    rc[1024 + c * 10 + t] = b2p[t];
  }
}

// Fold all rc-linear terms + biases into per-(b,o) additive vectors; writes u2 directly.
__global__ void k_rcgemv(const float* __restrict__ Wrc0,   const float* __restrict__ Brpt0,
                         const float* __restrict__ Brc0,   const float* __restrict__ Wrcb0,
                         const float* __restrict__ Bb0,    const float* __restrict__ Brcb0,
                         const float* __restrict__ Wrcl1,  const float* __restrict__ Bl1,
                         const float* __restrict__ Brcl1,  const float* __restrict__ Wrcbs1,
                         const float* __restrict__ Bbs1,   const float* __restrict__ Brcbs1,
                         const float* __restrict__ Wrcfin, const float* __restrict__ Bfincpt,
                         const float* __restrict__ Brcfin, const float* __restrict__ Wrcbfin,
                         const float* __restrict__ Brcbfin, float* __restrict__ ws,
                         float* __restrict__ out) {
  int tid = blockIdx.x * blockDim.x + threadIdx.x;  // 9984 = B*1248
  int b = tid / 1248, r = tid % 1248;
  const float* rc = ws + WS_RC + (long)b * NRCQ;
  const float* wrow;
  float bias;
  float* dst;
  if (r < 512)      { int o = r;       wrow = Wrc0   + (long)o * NRCQ; bias = Brpt0[o] + Brc0[o];    dst = ws + WS_ADD0 + b * 512 + o; }
  else if (r < 544) { int o = r - 512; wrow = Wrcb0  + (long)o * NRCQ; bias = Bb0[o] + Brcb0[o];     dst = ws + WS_ADDB0 + b * 32 + o; }
  else if (r < 576) { int o = r - 544; wrow = Wrcl1  + (long)o * NRCQ; bias = Bl1[o] + Brcl1[o];     dst = ws + WS_ADDL1 + b * 32 + o; }
  else if (r < 608) { int o = r - 576; wrow = Wrcbs1 + (long)o * NRCQ; bias = Bbs1[o] + Brcbs1[o];   dst = ws + WS_ADDBIAS1 + b * 32 + o; }
  else if (r < 928) { int o = r - 608; wrow = Wrcfin + (long)o * NRCQ; bias = Bfincpt[o] + Brcfin[o]; dst = ws + WS_ADDFIN + b * 320 + o; }
  else              { int o = r - 928; wrow = Wrcbfin + (long)o * NRCQ; bias = Brcbfin[o];           dst = out + U2_OFF + b * 320 + o; }
  float acc = bias;
  for (int i = 0; i < NRCQ; ++i) acc += wrow[i] * rc[i];
  *dst = acc;
}

// ---------------- Phase 2: WMMA GEMMs ----------------

// z0 = W_w0_rpt(512x576) @ rpt + add0 -> zw0 with (co, d, nin) permuted store
__global__ void k_gemm_zw0(const float* __restrict__ Wrpt, const float* __restrict__ ws,
                           float* __restrict__ out) {
  int wid = blockIdx.x * 8 + (threadIdx.x >> 5), lane = threadIdx.x & 31;
  int nt = wid & 31, mt = (wid >> 5) & 31, b = wid >> 10;
  const float* Bm = ws + WS_ROWCAT + (long)b * 608 * 512;
  const float* add0 = ws + WS_ADD0;
  v8f c = wmma_tile_f32<0>(Wrpt, 576, Bm, 512, 576, mt * 16, nt * 16, lane);
  int l15 = lane & 15, hi = lane >> 4, d = nt * 16 + l15;
#pragma unroll
  for (int r = 0; r < 8; ++r) {
    int o = mt * 16 + r + 8 * hi;
    int co = o >> 4, nin = o & 15;
    float v = c[r] + add0[b * 512 + o];
    __builtin_nontemporal_store(v, out + ZW0_OFF + ((long)(b * 32 + co) * 512 + d) * 16 + nin);
  }
}

// u0 = W_b0(32x576) @ rpt + addb0
__global__ void k_gemm_u0(const float* __restrict__ Wb0, const float* __restrict__ ws,
                          float* __restrict__ out) {
  int wid = blockIdx.x * 8 + (threadIdx.x >> 5), lane = threadIdx.x & 31;
  int nt = wid & 31, mt = (wid >> 5) & 1, b = wid >> 6;
  const float* Bm = ws + WS_ROWCAT + (long)b * 608 * 512;
  const float* addb0 = ws + WS_ADDB0;
  v8f c = wmma_tile_f32<0>(Wb0, 576, Bm, 512, 576, mt * 16, nt * 16, lane);
  int l15 = lane & 15, hi = lane >> 4, d = nt * 16 + l15;
#pragma unroll
  for (int r = 0; r < 8; ++r) {
    int o = mt * 16 + r + 8 * hi;
    out[U0_OFF + (long)(b * 32 + o) * 512 + d] = c[r] + addb0[b * 32 + o];
  }
}

// rowv = W_l1_r(perm, 32x608) @ rowcat + B_l1_r -> ws
__global__ void k_gemm_rowv(const float* __restrict__ Wl1r, const float* __restrict__ Bl1r,
                            float* __restrict__ ws) {
  int wid = blockIdx.x * 8 + (threadIdx.x >> 5), lane = threadIdx.x & 31;
  int nt = wid & 31, mt = (wid >> 5) & 1, b = wid >> 6;
  const float* Bm = ws + WS_ROWCAT + (long)b * 608 * 512;
  v8f c = wmma_tile_f32<1>(Wl1r, 608, Bm, 512, 608, mt * 16, nt * 16, lane);
  int l15 = lane & 15, hi = lane >> 4, d = nt * 16 + l15;
#pragma unroll
  for (int r = 0; r < 8; ++r) {
    int o = mt * 16 + r + 8 * hi;
    ws[WS_ROWV + (long)(b * 32 + o) * 512 + d] = c[r] + Bl1r[o];
  }
}

// colv = W_l1_c(perm, 32x416) @ colcat + B_l1_c -> ws
__global__ void k_gemm_colv(const float* __restrict__ Wl1c, const float* __restrict__ Bl1c,
                            float* __restrict__ ws) {
  int wid = blockIdx.x * 8 + (threadIdx.x >> 5), lane = threadIdx.x & 31;
  int nt = wid & 31, mt = (wid >> 5) & 1, b = wid >> 6;
  const float* Bm = ws + WS_COLCAT + (long)b * 416 * 512;
  v8f c = wmma_tile_f32<2>(Wl1c, 416, Bm, 512, 416, mt * 16, nt * 16, lane);
  int l15 = lane & 15, hi = lane >> 4, d = nt * 16 + l15;
#pragma unroll
  for (int r = 0; r < 8; ++r) {
    int o = mt * 16 + r + 8 * hi;
    ws[WS_COLV + (long)(b * 32 + o) * 512 + d] = c[r] + Bl1c[o];
  }
}

// u1 = W_bias1(perm, 32x416) @ colcat + addbias1
__global__ void k_gemm_u1(const float* __restrict__ Wbs1, const float* __restrict__ ws,
                          float* __restrict__ out) {
  int wid = blockIdx.x * 8 + (threadIdx.x >> 5), lane = threadIdx.x & 31;
  int nt = wid & 31, mt = (wid >> 5) & 1, b = wid >> 6;
  const float* Bm = ws + WS_COLCAT + (long)b * 416 * 512;
  const float* addbs1 = ws + WS_ADDBIAS1;
  v8f c = wmma_tile_f32<2>(Wbs1, 416, Bm, 512, 416, mt * 16, nt * 16, lane);
  int l15 = lane & 15, hi = lane >> 4, d = nt * 16 + l15;
#pragma unroll
  for (int r = 0; r < 8; ++r) {
    int o = mt * 16 + r + 8 * hi;
    out[U1_OFF + (long)(b * 32 + o) * 512 + d] = c[r] + addbs1[b * 32 + o];
  }
}

// zw2 = W_fin_cpt(320x352) @ cpt (= colcat rows 0..351) + addfin (contiguous store)
__global__ void k_gemm_zw2(const float* __restrict__ Wfin, const float* __restrict__ ws,
                           float* __restrict__ out) {
  int wid = blockIdx.x * 8 + (threadIdx.x >> 5), lane = threadIdx.x & 31;
  int nt = wid % 32, mtb = wid / 32, mt = mtb % 20, b = mtb / 20;
  const float* Bm = ws + WS_COLCAT + (long)b * 416 * 512;
  const float* addfin = ws + WS_ADDFIN;
  v8f c = wmma_tile_f32<0>(Wfin, 352, Bm, 512, 352, mt * 16, nt * 16, lane);
  int l15 = lane & 15, hi = lane >> 4, d = nt * 16 + l15;
#pragma unroll
  for (int r = 0; r < 8; ++r) {
    int o = mt * 16 + r + 8 * hi;
    out[ZW2_OFF + (long)(b * 320 + o) * 512 + d] = c[r] + addfin[b * 320 + o];
  }
}

// zw1[b,o,h,w] = (W_l1 @ w1[b,:,h,w])[o] + addl1[b,o] + rowv[b,o,w] + colv[b,o,h]
// Dual M-tile per wave: w1 streamed exactly once (NT), both 16-row halves share B.
__global__ void k_gemm_zw1(const float* __restrict__ w1, const float* __restrict__ Wl1,
                           const float* __restrict__ ws, float* __restrict__ out) {
  int wid = blockIdx.x * 8 + (threadIdx.x >> 5), lane = threadIdx.x & 31;
  int nt = wid & 31, h = (wid >> 5) & 511, b = wid >> 14;
  const int l15 = lane & 15, hi = lane >> 4, kh = hi << 1;
  const long ldb = (long)D2Q * D1Q;
  const float* bcol  = w1 + (long)b * CQ * D2Q * D1Q + (long)h * D1Q + nt * 16 + l15;
  const float* arow0 = Wl1 + l15 * 32;         // M rows 0..15
  const float* arow1 = Wl1 + (16 + l15) * 32;  // M rows 16..31
  v8f acc0 = {0.f, 0.f, 0.f, 0.f, 0.f, 0.f, 0.f, 0.f};
  v8f acc1 = {0.f, 0.f, 0.f, 0.f, 0.f, 0.f, 0.f, 0.f};
#pragma unroll
  for (int k = 0; k < 32; k += 4) {
    int ka = k + kh;
    v2f a0, a1, bv;
    a0.x = arow0[ka];
    a0.y = arow0[ka + 1];
    a1.x = arow1[ka];
    a1.y = arow1[ka + 1];
    bv.x = __builtin_nontemporal_load(bcol + (long)ka * ldb);
    bv.y = __builtin_nontemporal_load(bcol + (long)(ka + 1) * ldb);
    acc0 = __builtin_amdgcn_wmma_f32_16x16x4_f32(false, a0, false, bv, (short)0, acc0,
                                                 false, false);
    acc1 = __builtin_amdgcn_wmma_f32_16x16x4_f32(false, a1, false, bv, (short)0, acc1,
                                                 false, false);
  }
  const float* addl1 = ws + WS_ADDL1;
  const float* rowv  = ws + WS_ROWV;
  const float* colv  = ws + WS_COLV;
  int w = nt * 16 + l15;
#pragma unroll
  for (int r = 0; r < 8; ++r) {
    int o0 = r + 8 * hi;
    int o1 = 16 + r + 8 * hi;
    float v0 = acc0[r] + addl1[b * 32 + o0] + rowv[(long)(b * 32 + o0) * 512 + w] +
               colv[(long)(b * 32 + o0) * 512 + h];
    float v1 = acc1[r] + addl1[b * 32 + o1] + rowv[(long)(b * 32 + o1) * 512 + w] +
               colv[(long)(b * 32 + o1) * 512 + h];
    __builtin_nontemporal_store(v0, out + ZW1_OFF + ((long)(b * 32 + o0) * 512 + h) * 512 + w);
    __builtin_nontemporal_store(v1, out + ZW1_OFF + ((long)(b * 32 + o1) * 512 + h) * 512 + w);
  }
}

extern "C" void kernel_launch(void* const* d_in, const int* in_sizes, int n_in,
                              void* d_out, int out_size, void* d_ws, size_t ws_size,
                              hipStream_t stream) {
  const float* w0 = (const float*)d_in[0];
  const float* w1 = (const float*)d_in[1];
  const float* w2 = (const float*)d_in[2];
  const float* b0 = (const float*)d_in[3];
  const float* b1 = (const float*)d_in[4];
  const float* b2 = (const float*)d_in[5];
  const float* W_w0_rpt = (const float*)d_in[6];
  const float* B_w0_rpt = (const float*)d_in[7];
  const float* W_w0_rc  = (const float*)d_in[8];
  const float* B_w0_rc  = (const float*)d_in[9];
  const float* W_b0     = (const float*)d_in[10];
  const float* B_b0     = (const float*)d_in[11];
  const float* W_b0_rc  = (const float*)d_in[12];
  const float* B_b0_rc  = (const float*)d_in[13];
  const float* W_l1     = (const float*)d_in[14];
  const float* B_l1     = (const float*)d_in[15];
  const float* W_l1_rc  = (const float*)d_in[16];
  const float* B_l1_rc  = (const float*)d_in[17];
  const float* W_l1_r   = (const float*)d_in[18];
  const float* B_l1_r   = (const float*)d_in[19];
  const float* W_l1_c   = (const float*)d_in[20];
  const float* B_l1_c   = (const float*)d_in[21];
  const float* W_bias1  = (const float*)d_in[22];
  const float* B_bias1  = (const float*)d_in[23];
  const float* W_bias1_rc = (const float*)d_in[24];
  const float* B_bias1_rc = (const float*)d_in[25];
  const float* W_fin_cpt  = (const float*)d_in[26];
  const float* B_fin_cpt  = (const float*)d_in[27];
  const float* W_fin_rc   = (const float*)d_in[28];
  const float* B_fin_rc   = (const float*)d_in[29];
  const float* W_bfin_rc  = (const float*)d_in[30];
  const float* B_bfin_rc  = (const float*)d_in[31];
  float* out = (float*)d_out;
  float* ws  = (float*)d_ws;

  // Phase 1
  k_rowcat<<<512, 256, 0, stream>>>(w0, w1, b0, ws);
  k_colcat<<<512, 256, 0, stream>>>(w1, w2, b1, ws);
  k_rc<<<256, 256, 0, stream>>>(w0, w1, w2, b0, b1, b2, ws);
  k_rcgemv<<<39, 256, 0, stream>>>(W_w0_rc, B_w0_rpt, B_w0_rc, W_b0_rc, B_b0, B_b0_rc,
                                   W_l1_rc, B_l1, B_l1_rc, W_bias1_rc, B_bias1,
                                   B_bias1_rc, W_fin_rc, B_fin_cpt, B_fin_rc,
                                   W_bfin_rc, B_bfin_rc, ws, out);
  // Phase 2 (WMMA)
  k_gemm_zw0<<<1024, 256, 0, stream>>>(W_w0_rpt, ws, out);
  k_gemm_u0<<<64, 256, 0, stream>>>(W_b0, ws, out);
  k_gemm_rowv<<<64, 256, 0, stream>>>(W_l1_r, B_l1_r, ws);
  k_gemm_colv<<<64, 256, 0, stream>>>(W_l1_c, B_l1_c, ws);
  k_gemm_u1<<<64, 256, 0, stream>>>(W_bias1, ws, out);
  k_gemm_zw2<<<640, 256, 0, stream>>>(W_fin_cpt, ws, out);
  k_gemm_zw1<<<16384, 256, 0, stream>>>(w1, W_l1, ws, out);

  (void)in_sizes; (void)n_in; (void)out_size; (void)ws_size;
}